// BatchDynamicSoftLabelAssigner3D_78761110274664
// MI455X (gfx1250) — compile-verified
//
#include <hip/hip_runtime.h>
#include <hip/hip_bf16.h>
#include <math.h>

// BatchDynamicSoftLabelAssigner3D for MI455X (gfx1250, wave32).
// Pass 1: per (b,g) column stats over N (top-13 iou sum -> dynamic_k, 14 smallest
//         costs -> threshold), one wave per column, register selection networks
//         + LDS tree merge. Writes thr[b*G+g] to d_ws.
// Pass 2: 16(n)x64(g) tiles; pairwise xz dot via V_WMMA_F32_16X16X4_F32
//         (d^2 = |p|^2 + |g|^2 - 2 p.g), 4 unrolled WMMAs per wave, cost/iou
//         finished per element, staged in LDS, then per-row assignment logic.

typedef float v2f __attribute__((ext_vector_type(2)));
typedef float v8f __attribute__((ext_vector_type(8)));

#define NCLS 10
#define GDIM 64
#define INF_COST 1e8f

__device__ __forceinline__ float cost_formula(float iou, float s, float dist) {
  // iou_cost + cls_cost + soft_center_prior (reference semantics)
  float ic    = -logf(iou + 1e-7f) * 3.0f;
  float sig   = 1.0f / (1.0f + expf(-s));
  float scale = iou - sig;
  float bce   = fmaxf(s, 0.0f) - s * iou + log1pf(expf(-fabsf(s)));
  float soft  = expf((dist - 3.0f) * 2.302585092994046f); // 10^(dist-3)
  return bce * scale * scale + ic + soft;
}

// ---------------------------------------------------------------- pass 1 ----
__global__ __launch_bounds__(32) void pass1_thresh(
    const float* __restrict__ pred_bboxes, const float* __restrict__ pred_scores,
    const float* __restrict__ priors, const int* __restrict__ gt_labels,
    const float* __restrict__ gt_bboxes, const float* __restrict__ gt_center,
    const float* __restrict__ gt_b3d, const float* __restrict__ pad_flag,
    float* __restrict__ thr_out, int B, int N)
{
  const int bg   = blockIdx.x;          // b*GDIM + g
  const int b    = bg / GDIM;
  const int lane = threadIdx.x;

  const float x1  = gt_bboxes[(size_t)bg * 4 + 0];
  const float y1  = gt_bboxes[(size_t)bg * 4 + 1];
  const float x2  = gt_bboxes[(size_t)bg * 4 + 2];
  const float y2  = gt_bboxes[(size_t)bg * 4 + 3];
  const float gcx = gt_center[(size_t)bg * 2 + 0];
  const float gcy = gt_center[(size_t)bg * 2 + 1];
  const float gx  = gt_b3d[(size_t)bg * 7 + 0];
  const float gz  = gt_b3d[(size_t)bg * 7 + 2];
  const int   lab = gt_labels[bg];
  const float pad = pad_flag[bg];

  float ti[13];                         // top-13 ious, descending
  float tc[14];                         // 14 smallest costs, ascending
#pragma unroll
  for (int i = 0; i < 13; ++i) ti[i] = -1.0f;
#pragma unroll
  for (int i = 0; i < 14; ++i) tc[i] = 3.0e38f;

  for (int n = lane; n < N; n += 32) {
    const float cx = priors[n * 4 + 0];
    const float cy = priors[n * 4 + 1];
    const float sx = priors[n * 4 + 2];
    const float sy = priors[n * 4 + 3];
    const size_t bn = (size_t)b * N + n;
    const float px = pred_bboxes[bn * 7 + 0];
    const float pz = pred_bboxes[bn * 7 + 2];
    const float dx = px - gx, dz = pz - gz;
    const float d  = sqrtf(dx * dx + dz * dz);
    const float iou = fminf(fmaxf(1.0f - d * 0.25f, 0.0f), 1.0f);
    const bool ingts = (cx > x1) && (cy > y1) && (cx < x2) && (cy < y2) && (pad > 0.0f);
    float cost = INF_COST;
    if (ingts) {
      const float s  = pred_scores[bn * NCLS + lab];
      const float ux = (cx - gcx) / sx, uy = (cy - gcy) / sy;
      const float dist = sqrtf(ux * ux + uy * uy);
      cost = cost_formula(iou, s, dist);
    }
    float v = iou;                      // unrolled insertion -> stays in VGPRs
#pragma unroll
    for (int i = 0; i < 13; ++i) { if (v > ti[i]) { float t = ti[i]; ti[i] = v; v = t; } }
    float w = cost;
#pragma unroll
    for (int i = 0; i < 14; ++i) { if (w < tc[i]) { float t = tc[i]; tc[i] = w; w = t; } }
  }

  __shared__ float sI[32 * 13];
  __shared__ float sC[32 * 14];
  for (int off = 16; off >= 1; off >>= 1) {
#pragma unroll
    for (int i = 0; i < 13; ++i) sI[lane * 13 + i] = ti[i];
#pragma unroll
    for (int i = 0; i < 14; ++i) sC[lane * 14 + i] = tc[i];
    __syncthreads();
    if (lane < off) {
      for (int j = 0; j < 13; ++j) {
        float v = sI[(lane + off) * 13 + j];
#pragma unroll
        for (int i = 0; i < 13; ++i) { if (v > ti[i]) { float t = ti[i]; ti[i] = v; v = t; } }
      }
      for (int j = 0; j < 14; ++j) {
        float w = sC[(lane + off) * 14 + j];
#pragma unroll
        for (int i = 0; i < 14; ++i) { if (w < tc[i]) { float t = tc[i]; tc[i] = w; w = t; } }
      }
    }
    __syncthreads();
  }

  if (lane == 0) {
    float ssum = 0.0f;
#pragma unroll
    for (int i = 0; i < 13; ++i) ssum += ti[i];
    int k = (int)ssum;                  // trunc toward zero, like astype(int32)
    k = k < 1 ? 1 : (k > 13 ? 13 : k);
    thr_out[bg] = 0.5f * (tc[k] + tc[k - 1]);
  }
}

// ---------------------------------------------------------------- pass 2 ----
#define WAVES 4
__global__ __launch_bounds__(32 * WAVES) void pass2_assign(
    const float* __restrict__ pred_bboxes, const float* __restrict__ pred_scores,
    const float* __restrict__ priors, const int* __restrict__ gt_labels,
    const float* __restrict__ gt_bboxes, const float* __restrict__ gt_center,
    const float* __restrict__ gt_b3d, const float* __restrict__ pad_flag,
    const float* __restrict__ thr_in, float* __restrict__ out, int B, int N)
{
  const int tid  = threadIdx.x;
  const int wave = tid >> 5;
  const int lane = tid & 31;
  const int ntiles = N >> 4;
  const int blocksPerB = ntiles / WAVES;
  const int b = blockIdx.x / blocksPerB;
  const int n0 = ((blockIdx.x % blocksPerB) * WAVES + wave) * 16; // this wave's 16 priors

  // Padded with a trailing zero slot so lanes 16-31 (which must carry K=2,3 = 0
  // for the WMMA operands) can load unconditionally -> no exec save/restore.
  __shared__ float gL[64], gT[64], gR[64], gB[64];
  __shared__ float gCX[64], gCY[64], gGG[64], gPad[64], gThr[64];
  __shared__ float gXp[65], gZp[65];            // [64] = 0.0
  __shared__ int   gLab[64];
  __shared__ float pCX[WAVES][16], pCY[WAVES][16], pSX[WAVES][16], pSY[WAVES][16];
  __shared__ float pPP[WAVES][16];
  __shared__ float pXp[WAVES][17], pZp[WAVES][17]; // [16] = 0.0
  __shared__ float costT[WAVES][16][68];        // pad to 68: conflict-free column walks
  __shared__ float iouT[WAVES][16][68];

  for (int g = tid; g < 64; g += 32 * WAVES) {
    const size_t o = (size_t)b * 64 + g;
    gL[g]  = gt_bboxes[o * 4 + 0];  gT[g] = gt_bboxes[o * 4 + 1];
    gR[g]  = gt_bboxes[o * 4 + 2];  gB[g] = gt_bboxes[o * 4 + 3];
    gCX[g] = gt_center[o * 2 + 0];  gCY[g] = gt_center[o * 2 + 1];
    const float x = gt_b3d[o * 7 + 0], z = gt_b3d[o * 7 + 2];
    gXp[g] = x; gZp[g] = z; gGG[g] = x * x + z * z;
    gLab[g] = gt_labels[o]; gPad[g] = pad_flag[o]; gThr[g] = thr_in[o];
  }
  if (tid == 0) { gXp[64] = 0.0f; gZp[64] = 0.0f; }
  if (lane < 16) {
    const int n = n0 + lane;
    pCX[wave][lane] = priors[n * 4 + 0]; pCY[wave][lane] = priors[n * 4 + 1];
    pSX[wave][lane] = priors[n * 4 + 2]; pSY[wave][lane] = priors[n * 4 + 3];
    const size_t bn = (size_t)b * N + n;
    const float x = pred_bboxes[bn * 7 + 0], z = pred_bboxes[bn * 7 + 2];
    pXp[wave][lane] = x; pZp[wave][lane] = z; pPP[wave][lane] = x * x + z * z;
  }
  if (lane == 16) { pXp[wave][16] = 0.0f; pZp[wave][16] = 0.0f; }
  __syncthreads();

  // A operand: 16x4 f32, K=2,3 zero. Lanes 0-15: VGPR0=K0(px), VGPR1=K1(pz);
  // lanes 16-31 read the zero slot.
  const int  lidx = lane & 15;
  const bool loh  = lane < 16;
  const int  aidx = loh ? lidx : 16;
  v2f A;
  A.x = pXp[wave][aidx];
  A.y = pZp[wave][aidx];
  const int mbase = (lane >> 4) * 8;   // D layout: VGPR j -> row m = j + 8*(lane>=16)

#pragma unroll
  for (int t = 0; t < 4; ++t) {
    const int gcol = t * 16 + lidx;
    const int bidx = loh ? gcol : 64;  // zero slot for upper half-wave
    v2f Bm;
    Bm.x = gXp[bidx];
    Bm.y = gZp[bidx];
    v8f Cm = {0.f, 0.f, 0.f, 0.f, 0.f, 0.f, 0.f, 0.f};
    v8f D = __builtin_amdgcn_wmma_f32_16x16x4_f32(
        /*neg_a=*/false, A, /*neg_b=*/false, Bm,
        /*c_mod=*/(short)0, Cm, /*reuse_a=*/false, /*reuse_b=*/false);

    const float bx1 = gL[gcol], by1 = gT[gcol], bx2 = gR[gcol], by2 = gB[gcol];
    const float gcxv = gCX[gcol], gcyv = gCY[gcol], ggv = gGG[gcol], padv = gPad[gcol];
    const int   lab = gLab[gcol];
#pragma unroll
    for (int j = 0; j < 8; ++j) {
      const int m = mbase + j;
      const float dot = D[j];
      float d2 = pPP[wave][m] + ggv - 2.0f * dot;
      d2 = fmaxf(d2, 0.0f);
      const float d   = sqrtf(d2);
      const float iou = fminf(fmaxf(1.0f - d * 0.25f, 0.0f), 1.0f);
      const float cxv = pCX[wave][m], cyv = pCY[wave][m];
      const bool ingts = (cxv > bx1) && (cyv > by1) && (cxv < bx2) && (cyv < by2) && (padv > 0.0f);
      float cost = INF_COST;
      if (ingts) {
        const float s  = pred_scores[((size_t)b * N + n0 + m) * NCLS + lab];
        const float ux = (cxv - gcxv) / pSX[wave][m];
        const float uy = (cyv - gcyv) / pSY[wave][m];
        const float dist = sqrtf(ux * ux + uy * uy);
        cost = cost_formula(iou, s, dist);
      }
      costT[wave][m][gcol] = cost;
      iouT[wave][m][gcol]  = iou;
    }
  }
  __syncthreads();

  if (lane < 16) {
    const int m = lane;
    const int n = n0 + m;
    float bestC = 3.0e38f;  int amin = 0;     // argmin over full cost
    float bestM = 3.0e38f;  int aminM = 0;    // argmin over masked cost
    int cnt = 0, first = -1;
    unsigned long long mask = 0ull;
    for (int g = 0; g < 64; ++g) {
      const float c = costT[wave][m][g];
      const bool match = (c < 1e7f) && (c <= gThr[g]);   // c<1e7 <=> in_gts
      if (c < bestC) { bestC = c; amin = g; }
      if (match) {
        if (first < 0) first = g;
        ++cnt;
        if (c < bestM) { bestM = c; aminM = g; }
        mask |= (1ull << g);
      }
    }
    const bool multi = cnt > 1;
    if (multi) mask = (1ull << aminM);
    const bool matched = cnt > 0;
    const int gt_idx = multi ? aminM : (first >= 0 ? first : 0);
    const float metrics = matched ? iouT[wave][m][gt_idx] : iouT[wave][m][amin];
    int labl;
    if (matched)                 labl = gLab[gt_idx];
    else if (metrics < 1e-7f)    labl = NCLS;
    else                         labl = gLab[amin];

    const size_t BN   = (size_t)B * N;
    const size_t base = (size_t)b * N + n;
    out[base]          = (float)labl;   // assigned_labels
    out[BN + base]     = 1.0f;          // assigned_labels_weights
    out[2 * BN + base] = metrics;       // assign_metrics

    // matching: 64 floats per row, written as 16 x b128 stores
    float4* mo4 = (float4*)(out + 3 * BN + base * 64);
#pragma unroll 4
    for (int q = 0; q < 16; ++q) {
      const int g = q * 4;
      float4 v;
      v.x = ((mask >> (g + 0)) & 1ull) ? 1.0f : 0.0f;
      v.y = ((mask >> (g + 1)) & 1ull) ? 1.0f : 0.0f;
      v.z = ((mask >> (g + 2)) & 1ull) ? 1.0f : 0.0f;
      v.w = ((mask >> (g + 3)) & 1ull) ? 1.0f : 0.0f;
      mo4[q] = v;
    }
  }
}

// ---------------------------------------------------------------- launch ----
extern "C" void kernel_launch(void* const* d_in, const int* in_sizes, int n_in,
                              void* d_out, int out_size, void* d_ws, size_t ws_size,
                              hipStream_t stream) {
  const float* pred_bboxes = (const float*)d_in[0];
  const float* pred_scores = (const float*)d_in[1];
  const float* priors      = (const float*)d_in[2];
  const int*   gt_labels   = (const int*)  d_in[3];
  const float* gt_bboxes   = (const float*)d_in[4];
  const float* gt_center   = (const float*)d_in[5];
  const float* gt_b3d      = (const float*)d_in[6];
  const float* pad_flag    = (const float*)d_in[7];
  float* out = (float*)d_out;

  const int N = in_sizes[2] / 4;        // priors: (N,4)
  const int B = in_sizes[0] / (N * 7);  // pred_bboxes: (B,N,7); G fixed at 64

  float* thr_ws = (float*)d_ws;         // B*GDIM floats of scratch

  pass1_thresh<<<B * GDIM, 32, 0, stream>>>(
      pred_bboxes, pred_scores, priors, gt_labels, gt_bboxes, gt_center,
      gt_b3d, pad_flag, thr_ws, B, N);

  const int blocks = B * ((N / 16) / WAVES);
  pass2_assign<<<blocks, 32 * WAVES, 0, stream>>>(
      pred_bboxes, pred_scores, priors, gt_labels, gt_bboxes, gt_center,
      gt_b3d, pad_flag, thr_ws, out, B, N);
}